// KPConvFeatureExtractor_56831007261056
// MI455X (gfx1250) — compile-verified
//
#include <hip/hip_runtime.h>

// ---------------------------------------------------------------------------
// KPConv feature extractor for gfx1250 (MI455X).
//   out[n,d] = sum_{k,c} ( sum_h exp(-|p_nbr - p_n - kp_k|^2 / 2s^2) * p_nbr[c] ) * W[k,c,d] + bias[d]
//
// wave32 owns 16 points; half-waves split the 32 neighbors, merged via
// shfl_xor(16). Gaussian arg strength-reduced with per-kernel-point constants
// pinned to SGPRs via v_readfirstlane (scalar FMA operands, zero VGPR cost):
//   ES*d2 = ES*|rel|^2 + (ES*|kp_k|^2) + rel . (-2*ES*kp_k)
// Neighbor loop: outer quad loop kept ROLLED (4 iters) to bound register
// pressure / code size; inner 4 neighbors x 15 kernel points unrolled for ILP.
// Final (16x48)x(48x32) contraction: 24x chained V_WMMA_F32_16X16X4_F32,
// B fragments from LDS (weights staged transposed+padded once per block).
// ---------------------------------------------------------------------------

#define KPTS  15
#define HNBR  32
#define COUT  32
#define NPTS  65536

#define KC    45      // KPTS * CIN
#define KCPAD 48      // padded to WMMA K-step multiple
#define NSTEP 12      // KCPAD / 4

// exp(-d2/(2*sigma^2)) = exp2(d2 * ES), ES = -log2(e)/(2*0.05^2)
#define EXP_SCALE (-288.53900817779268f)

typedef __attribute__((ext_vector_type(2))) float v2f;
typedef __attribute__((ext_vector_type(8))) float v8f;
typedef __attribute__((ext_vector_type(4))) int   v4i;

// Pin a wave-uniform float into an SGPR.
__device__ __forceinline__ float uniform_f(float v) {
    return __int_as_float(__builtin_amdgcn_readfirstlane(__float_as_int(v)));
}

__global__ __launch_bounds__(256, 1) void kpconv_wmma_kernel(
    const float* __restrict__ points,   // (N,3)
    const int*   __restrict__ nbr,      // (N,H) int32
    const float* __restrict__ kp,       // (K,3)
    const float* __restrict__ W,        // (45,32) row-major
    const float* __restrict__ bias,     // (32)
    float*       __restrict__ out,      // (N,32)
    int nTiles)
{
    // Transposed + zero-padded weights: Wt[d][kc], d in [0,32), kc in [0,48)
    __shared__ float Wt[COUT * KCPAD];

    const int tid = threadIdx.x;
#pragma unroll
    for (int e = tid; e < COUT * KCPAD; e += 256) {
        const int d = e / KCPAD, kc = e % KCPAD;
        const int kcc = (kc < KC) ? kc : (KC - 1);      // clamp: no branch
        const float v = W[kcc * COUT + d];              // unconditional load
        Wt[e] = (kc < KC) ? v : 0.0f;                   // cndmask
    }
    __syncthreads();

    const int lane  = tid & 31;
    const int half  = lane >> 4;        // which 16-lane half of the wave
    const int col   = lane & 15;        // point-in-tile / output column
    const int wave  = (blockIdx.x * blockDim.x + tid) >> 5;
    const int nWave = (gridDim.x * blockDim.x) >> 5;

    // Folded Gaussian constants, pinned to SGPRs (uniform, constant-indexed).
    float ekx[KPTS], eky[KPTS], ekz[KPTS], ec[KPTS];
#pragma unroll
    for (int k = 0; k < KPTS; ++k) {
        const float x = kp[k * 3 + 0];
        const float y = kp[k * 3 + 1];
        const float z = kp[k * 3 + 2];
        ekx[k] = uniform_f(-2.0f * EXP_SCALE * x);
        eky[k] = uniform_f(-2.0f * EXP_SCALE * y);
        ekz[k] = uniform_f(-2.0f * EXP_SCALE * z);
        ec[k]  = uniform_f(EXP_SCALE * (x * x + y * y + z * z));
    }
    const float b0 = bias[col];
    const float b1 = bias[16 + col];

    for (int tile = wave; tile < nTiles; tile += nWave) {
        const int m = tile * 16 + col;          // this lane's point
        const float* pc = points + m * 3;
        const float cx = pc[0], cy = pc[1], cz = pc[2];   // -> global_load_b96

        const v4i* ip = (const v4i*)(nbr + m * HNBR + half * 16);

        float pw[KCPAD];                        // weighted row (partial)
#pragma unroll
        for (int j = 0; j < KCPAD; ++j) pw[j] = 0.0f;

        // ROLLED quad loop: bounds code size + register pressure; each
        // iteration front-loads 1x b128 (indices) + 4x b96 (coords), then
        // 4*15 = 60 exp evaluations of VALU to hide them.
#pragma unroll 1
        for (int q = 0; q < 4; ++q) {
            const v4i iv = ip[q];               // global_load_b128
            float nx[4], ny[4], nz[4];
#pragma unroll
            for (int u = 0; u < 4; ++u) {
                int idx = iv[u];
                idx = (idx < 0) ? 0 : ((idx >= NPTS) ? NPTS - 1 : idx);
                const float* pn = points + idx * 3;   // -> global_load_b96
                nx[u] = pn[0]; ny[u] = pn[1]; nz[u] = pn[2];
            }
#pragma unroll
            for (int u = 0; u < 4; ++u) {
                const float rx = nx[u] - cx, ry = ny[u] - cy, rz = nz[u] - cz;
                float rs = rx * rx;
                rs = fmaf(ry, ry, rs);
                rs = fmaf(rz, rz, rs);
                rs *= EXP_SCALE;                       // ES*|rel|^2
#pragma unroll
                for (int k = 0; k < KPTS; ++k) {
                    float t = rs + ec[k];              // SGPR operand
                    t = fmaf(rx, ekx[k], t);
                    t = fmaf(ry, eky[k], t);
                    t = fmaf(rz, ekz[k], t);
                    const float w = __builtin_amdgcn_exp2f(t);  // v_exp_f32
                    pw[3 * k + 0] = fmaf(w, nx[u], pw[3 * k + 0]);
                    pw[3 * k + 1] = fmaf(w, ny[u], pw[3 * k + 1]);
                    pw[3 * k + 2] = fmaf(w, nz[u], pw[3 * k + 2]);
                }
            }
        }

        // merge the two neighbor halves: lane L <-> L^16 (permute, no barrier)
#pragma unroll
        for (int j = 0; j < KC; ++j)
            pw[j] += __shfl_xor(pw[j], 16, 32);
        // pw[45..47] stay 0 in all lanes (padding)

        // --- GEMM: weighted(16x48) x W(48x32), 24x V_WMMA_F32_16X16X4_F32 ---
        v8f acc0 = {};
        v8f acc1 = {};
        const float* wrow0 = Wt + col * KCPAD;          // d = col      (t=0)
        const float* wrow1 = Wt + (16 + col) * KCPAD;   // d = 16+col   (t=1)
#pragma unroll
        for (int s = 0; s < NSTEP; ++s) {
            // A 16x4: lanes 0-15 give K=4s..4s+1, lanes 16-31 give 4s+2..4s+3
            v2f a;
            a.x = half ? pw[4 * s + 2] : pw[4 * s + 0];
            a.y = half ? pw[4 * s + 3] : pw[4 * s + 1];
            // B 4x16 fragment from LDS: one ds_load_b64 per tile (8B aligned)
            const v2f bf0 = *(const v2f*)(wrow0 + 4 * s + 2 * half);
            const v2f bf1 = *(const v2f*)(wrow1 + 4 * s + 2 * half);
            acc0 = __builtin_amdgcn_wmma_f32_16x16x4_f32(
                false, a, false, bf0, (short)0, acc0, false, false);
            acc1 = __builtin_amdgcn_wmma_f32_16x16x4_f32(
                false, a, false, bf1, (short)0, acc1, false, false);
        }

        // --- store D: VGPR r holds row M = r + 8*half, N = col ---
        const int rowBase = tile * 16 + 8 * half;
#pragma unroll
        for (int r = 0; r < 8; ++r) {
            const int o = (rowBase + r) * COUT;
            out[o + col]      = acc0[r] + b0;
            out[o + 16 + col] = acc1[r] + b1;
        }
    }
}

extern "C" void kernel_launch(void* const* d_in, const int* in_sizes, int n_in,
                              void* d_out, int out_size, void* d_ws, size_t ws_size,
                              hipStream_t stream) {
    const float* points = (const float*)d_in[0];
    const int*   nbr    = (const int*)  d_in[1];   // jnp.int64 w/o x64 -> int32
    const float* kp     = (const float*)d_in[2];
    const float* W      = (const float*)d_in[3];
    const float* bias   = (const float*)d_in[4];
    float*       out    = (float*)d_out;

    const int nTiles = NPTS / 16;                   // 4096 point-tiles
    // 512 blocks x 8 waves = 4096 waves -> exactly 1 tile per wave.
    kpconv_wmma_kernel<<<dim3(512), dim3(256), 0, stream>>>(
        points, nbr, kp, W, bias, out, nTiles);
}